// HypergraphMotifConvE_40054865003218
// MI455X (gfx1250) — compile-verified
//
#include <hip/hip_runtime.h>
#include <hip/hip_bf16.h>

// ---------------------------------------------------------------------------
// HypergraphMotifConv pipeline for MI455X (gfx1250, wave32, WMMA + async LDS).
// ---------------------------------------------------------------------------

#define N_NODES 256
#define N_EDGES 65536
#define N_MOTIF 65536
#define C1 256
#define C2 256
#define C3 128
#define COUT 64
#define NNZ 524288
#define EE_NNZ 524288

#define KDIM 256            // K for all staged GEMMs
#define KSTRIDE (KDIM + 4)  // padded LDS row stride for A (bank-conflict-free)

typedef float v2f __attribute__((ext_vector_type(2)));
typedef float v8f __attribute__((ext_vector_type(8)));

__device__ __forceinline__ float leaky(float v) { return v >= 0.0f ? v : 0.01f * v; }

// Raw LDS byte offset of a pointer into a __shared__ object.
__device__ __forceinline__ unsigned lds_off(const void* p) {
  return (unsigned)(unsigned long long)(const __attribute__((address_space(3))) void*)p;
}

// ---------------------------------------------------------------------------
// Staged fp32 WMMA GEMM (K fixed = 256): C[M, 64*NT] = act(A @ B + bias)
//   * 128-thread block = 4 waves; block owns a 16-row slab of A and ALL
//     64*NT output columns (grid.x == 1).
//   * A panel (16 x 256 f32) staged once via global_load_async_to_lds_b128.
//   * B staged in 16-row K-chunks, double-buffered with async-to-LDS DMA:
//     prefetch chunk c+1 while computing chunk c; the inner loop is pure
//     ds_load + v_wmma_f32_16x16x4_f32 (no per-WMMA loadcnt waits).
// Fragment layouts (CDNA5 ISA 7.12.2, 32-bit):
//   A 16x4:  lane = 16*hi + m, a[j] = A[m][2*hi + j]
//   B 4x16:  lane = 16*hi + n, b[j] = B[2*hi + j][n]
//   C 16x16: acc[r] = C[r + 8*hi][lane & 15]
// EPI: 0 = none, 1 = leaky_relu.
// ---------------------------------------------------------------------------
template <int EPI, int NT>
__global__ __launch_bounds__(128) void gemm_wmma_staged(
    const float* __restrict__ A, const float* __restrict__ B,
    const float* __restrict__ bias, float* __restrict__ C) {
  constexpr int NDIM = 64 * NT;       // output width handled by the block
  constexpr int BSTRIDE = NDIM + 16;  // padded LDS row stride for B chunks
  constexpr int NCHUNK = KDIM / 16;   // 16 K-chunks of 16 rows each

  __shared__ float sA[16 * KSTRIDE];
  __shared__ float sB[2 * 16 * BSTRIDE];

  const int t = threadIdx.x;       // 0..127
  const int m0 = blockIdx.y * 16;  // A-row slab

  // ---- async-stage A[m0:m0+16, 0:256] into LDS ----
  {
    const float* gA = A + (size_t)m0 * KDIM;
#pragma unroll
    for (int r = 0; r < 8; ++r) {
      const int idx4 = r * 128 + t;       // 16-byte chunk id, 0..1023
      const int row = idx4 >> 6;          // 64 chunks per 256-float row
      const int colf = (idx4 & 63) << 2;  // float column
      const unsigned lds = lds_off(&sA[row * KSTRIDE + colf]);
      const unsigned goff = (unsigned)((row * KDIM + colf) * 4);
      asm volatile("global_load_async_to_lds_b128 %0, %1, %2"
                   :: "v"(lds), "v"(goff), "s"(gA)
                   : "memory");
    }
  }

  // ---- async-stage one 16-row chunk of B into buffer `buf` ----
  auto stage_b = [&](int chunk, int buf) {
#pragma unroll
    for (int r = 0; r < 2 * NT; ++r) {
      const int j = r * 128 + t;              // b128 chunk id, 0..256*NT-1
      const int row = j / (16 * NT);          // 16*NT b128 chunks per row
      const int colf = (j % (16 * NT)) << 2;  // float column
      const unsigned lds = lds_off(&sB[(buf * 16 + row) * BSTRIDE + colf]);
      const unsigned goff = (unsigned)(((chunk * 16 + row) * NDIM + colf) * 4);
      asm volatile("global_load_async_to_lds_b128 %0, %1, %2"
                   :: "v"(lds), "v"(goff), "s"(B)
                   : "memory");
    }
  };

  stage_b(0, 0);
  asm volatile("s_wait_asynccnt 0x0" ::: "memory");
  __syncthreads();

  // ---- compute: wave w owns columns [w*16*NT, (w+1)*16*NT) ----
  const int wave = t >> 5;
  const int lane = t & 31;
  const int lo = lane & 15;
  const int hi = lane >> 4;
  const int nbase = wave * (16 * NT);

  v8f acc[NT];
#pragma unroll
  for (int nt = 0; nt < NT; ++nt) {
    const float bv = bias ? bias[nbase + nt * 16 + lo] : 0.0f;
#pragma unroll
    for (int r = 0; r < 8; ++r) acc[nt][r] = bv;
  }

  for (int c = 0; c < NCHUNK; ++c) {
    if (c + 1 < NCHUNK) stage_b(c + 1, (c + 1) & 1);  // prefetch next chunk

    const float* bB = &sB[(c & 1) * 16 * BSTRIDE];
    const int kb = c * 16;
#pragma unroll
    for (int kk = 0; kk < 16; kk += 4) {
      const int kl = kk + 2 * hi;
      v2f a;
      a[0] = sA[lo * KSTRIDE + kb + kl];
      a[1] = sA[lo * KSTRIDE + kb + kl + 1];
#pragma unroll
      for (int nt = 0; nt < NT; ++nt) {
        const int n = nbase + nt * 16 + lo;
        v2f b;
        b[0] = bB[kl * BSTRIDE + n];
        b[1] = bB[(kl + 1) * BSTRIDE + n];
        acc[nt] = __builtin_amdgcn_wmma_f32_16x16x4_f32(
            false, a, false, b, (short)0, acc[nt], false, false);
      }
    }

    asm volatile("s_wait_asynccnt 0x0" ::: "memory");  // prefetch landed
    __syncthreads();                                   // all waves done w/ buffers
  }

#pragma unroll
  for (int nt = 0; nt < NT; ++nt) {
#pragma unroll
    for (int r = 0; r < 8; ++r) {
      float v = acc[nt][r];
      if (EPI == 1) v = leaky(v);
      C[(size_t)(m0 + r + 8 * hi) * NDIM + (nbase + nt * 16 + lo)] = v;
    }
  }
}

// ---------------------------------------------------------------------------
// Simple fp32 WMMA GEMM (one wave per 16x16 tile) for the odd shapes:
// TRANSA (y^T y) and K=128 (W_lin3 @ W_out).
// ---------------------------------------------------------------------------
template <int EPI, bool TRANSA>
__global__ void gemm_wmma_f32(const float* __restrict__ A,
                              const float* __restrict__ B,
                              const float* __restrict__ bias,
                              float* __restrict__ C,
                              int Mdim, int K, int Ndim) {
  const int lane = threadIdx.x & 31;
  const int lo = lane & 15;
  const int hi = lane >> 4;
  const int n0 = blockIdx.x * 16;
  const int m0 = blockIdx.y * 16;

  const float bv = bias ? bias[n0 + lo] : 0.0f;
  v8f acc;
#pragma unroll
  for (int r = 0; r < 8; ++r) acc[r] = bv;

  for (int k = 0; k < K; k += 4) {
    const int ka = k + 2 * hi;
    v2f a, b;
    if (TRANSA) {
      a[0] = A[(size_t)ka * Mdim + (m0 + lo)];
      a[1] = A[(size_t)(ka + 1) * Mdim + (m0 + lo)];
    } else {
      const float* ap = A + (size_t)(m0 + lo) * K + ka;
      a[0] = ap[0];
      a[1] = ap[1];
    }
    b[0] = B[(size_t)ka * Ndim + (n0 + lo)];
    b[1] = B[(size_t)(ka + 1) * Ndim + (n0 + lo)];
    acc = __builtin_amdgcn_wmma_f32_16x16x4_f32(
        false, a, false, b, (short)0, acc, false, false);
  }

#pragma unroll
  for (int r = 0; r < 8; ++r) {
    float v = acc[r];
    if (EPI == 1) v = leaky(v);
    C[(size_t)(m0 + r + 8 * hi) * Ndim + (n0 + lo)] = v;
  }
}

// ---------------------------------------------------------------------------
// Scatter / gather / segment helpers
// ---------------------------------------------------------------------------
__global__ void k_fill(float* __restrict__ p, float v, int n) {
  int i = blockIdx.x * blockDim.x + threadIdx.x;
  if (i < n) p[i] = v;
}

__global__ void k_count(const int* __restrict__ idx, float* __restrict__ cnt, int nnz) {
  int i = blockIdx.x * blockDim.x + threadIdx.x;
  if (i < nnz) atomicAdd(&cnt[idx[i]], 1.0f);
}

// out[dst[i]][c] += tab[src[i]][c]   (tab is a small row table)
__global__ void k_scatter_table(const float* __restrict__ tab,
                                const int* __restrict__ src,
                                const int* __restrict__ dst,
                                float* __restrict__ out, int C) {
  const int i = blockIdx.x;
  const int c = threadIdx.x;
  const int s = src[i];
  const int d = dst[i];
  atomicAdd(&out[(size_t)d * C + c], tab[(size_t)s * C + c]);
}

// row-wise multiply by 1/cnt (0 if cnt==0)
__global__ void k_scale_rows(float* __restrict__ out, const float* __restrict__ cnt, int C) {
  const int e = blockIdx.x;
  const int c = threadIdx.x;
  const float d = cnt[e];
  const float inv = d > 0.0f ? 1.0f / d : 0.0f;
  out[(size_t)e * C + c] *= inv;
}

// y[src[i]][c] += oute[he[i]][c]
__global__ void k_gather_nodes(const float* __restrict__ oute,
                               const int* __restrict__ src,
                               const int* __restrict__ he,
                               float* __restrict__ y, int C) {
  const int i = blockIdx.x;
  const int c = threadIdx.x;
  atomicAdd(&y[(size_t)src[i] * C + c], oute[(size_t)he[i] * C + c]);
}

// y = leaky(y * d_inv + b_hg)
__global__ void k_node_epi(float* __restrict__ y, const float* __restrict__ dcnt,
                           const float* __restrict__ bias, int C) {
  const int n = blockIdx.x;
  const int c = threadIdx.x;
  const float d = dcnt[n];
  const float inv = d > 0.0f ? 1.0f / d : 0.0f;
  y[(size_t)n * C + c] = leaky(y[(size_t)n * C + c] * inv + bias[c]);
}

// acc[col][c] += dis[row]*dis[col]*xw[row][c]
__global__ void k_gcn_agg(const float* __restrict__ xw,
                          const int* __restrict__ row,
                          const int* __restrict__ col,
                          const float* __restrict__ deg,
                          float* __restrict__ acc, int C) {
  const int i = blockIdx.x;
  const int c = threadIdx.x;
  const int r = row[i];
  const int cl = col[i];
  const float dr = deg[r];
  const float dc = deg[cl];
  const float norm = (dr > 0.0f ? rsqrtf(dr) : 0.0f) * (dc > 0.0f ? rsqrtf(dc) : 0.0f);
  atomicAdd(&acc[(size_t)cl * C + c], norm * xw[(size_t)r * C + c]);
}

// self-loop term + bias: acc[e][c] += (1/deg[e]) * xw[e][c] + b_gcn[c]
__global__ void k_gcn_epi(float* __restrict__ acc, const float* __restrict__ xw,
                          const float* __restrict__ deg, const float* __restrict__ bias,
                          int C) {
  const int e = blockIdx.x;
  const int c = threadIdx.x;
  const float d = deg[e];
  const float inv = d > 0.0f ? 1.0f / d : 0.0f;  // dis[e]^2
  acc[(size_t)e * C + c] += inv * xw[(size_t)e * C + c] + bias[c];
}

// segment_min with exactly 3 consecutive members per motif
__global__ void k_motif_min(const float* __restrict__ y, const int* __restrict__ me,
                            float* __restrict__ m, int C) {
  const int mid = blockIdx.x;
  const int c = threadIdx.x;
  const int e0 = me[3 * mid + 0];
  const int e1 = me[3 * mid + 1];
  const int e2 = me[3 * mid + 2];
  const float v = fminf(y[(size_t)e0 * C + c],
                        fminf(y[(size_t)e1 * C + c], y[(size_t)e2 * C + c]));
  m[(size_t)mid * C + c] = v;
}

// bc[j] = b_out[j] + sum_k b_lin3[k] * W_out[k][j]
__global__ void k_fuse_bias(const float* __restrict__ b3, const float* __restrict__ Wout,
                            const float* __restrict__ bout, float* __restrict__ bc) {
  const int j = threadIdx.x;  // 64
  float s = bout[j];
  for (int k = 0; k < C3; ++k) s += b3[k] * Wout[k * COUT + j];
  bc[j] = s;
}

// ---------------------------------------------------------------------------
extern "C" void kernel_launch(void* const* d_in, const int* in_sizes, int n_in,
                              void* d_out, int out_size, void* d_ws, size_t ws_size,
                              hipStream_t stream) {
  (void)in_sizes; (void)n_in; (void)out_size; (void)ws_size;

  const float* node_embeds = (const float*)d_in[0];
  const float* W_hg   = (const float*)d_in[1];
  const float* b_hg   = (const float*)d_in[2];
  const float* W_lin  = (const float*)d_in[3];
  const float* b_lin  = (const float*)d_in[4];
  const float* W_gcn  = (const float*)d_in[5];
  const float* b_gcn  = (const float*)d_in[6];
  const float* W_lin2 = (const float*)d_in[7];
  const float* b_lin2 = (const float*)d_in[8];
  const float* W_lin3 = (const float*)d_in[9];
  const float* b_lin3 = (const float*)d_in[10];
  const float* W_out  = (const float*)d_in[11];
  const float* b_out  = (const float*)d_in[12];
  const int* edge_index   = (const int*)d_in[13];  // [2, NNZ]
  const int* ee_index     = (const int*)d_in[14];  // [2, EE_NNZ]
  const int* motif_index  = (const int*)d_in[15];  // [2, 3*M]

  const int* src = edge_index;            // row 0
  const int* he  = edge_index + NNZ;      // row 1
  const int* ee_row = ee_index;           // row 0
  const int* ee_col = ee_index + EE_NNZ;  // row 1
  const int* m_edge = motif_index;        // row 0 (row 1 == repeat(arange(M),3))

  float* out = (float*)d_out;             // [M, COUT]

  // ---- workspace carve-up ----
  float* ws = (float*)d_ws;
  float* bufA  = ws;                              // E*C1 (out_e -> xw -> y2e)
  float* bufB  = bufA + (size_t)N_EDGES * C1;     // E*C1 (y_edge -> y_gcn -> motif m)
  float* s_x   = bufB + (size_t)N_EDGES * C1;     // [N, C1]
  float* s_y   = s_x   + N_NODES * C1;            // [N, C1]
  float* s_yty = s_y   + N_NODES * C1;            // [C1, C1]
  float* s_z   = s_yty + C1 * C1;                 // [C1, C1]
  float* s_Wc  = s_z   + C1 * C1;                 // [C2, COUT]
  float* s_bc  = s_Wc  + C2 * COUT;               // [COUT]
  float* f_d   = s_bc  + COUT;                    // [N] node degree
  float* f_b   = f_d   + N_NODES;                 // [E] hyperedge degree
  float* f_dee = f_b   + N_EDGES;                 // [E] GCN degree (incl. self loop)

  const size_t bigBytes = (size_t)N_EDGES * C1 * sizeof(float);

  // ---- init accumulators / degrees ----
  hipMemsetAsync(bufA, 0, bigBytes, stream);
  hipMemsetAsync(bufB, 0, bigBytes, stream);
  hipMemsetAsync(s_y, 0, (size_t)N_NODES * C1 * sizeof(float), stream);
  hipMemsetAsync(f_d, 0, N_NODES * sizeof(float), stream);
  hipMemsetAsync(f_b, 0, N_EDGES * sizeof(float), stream);
  k_fill<<<(N_EDGES + 255) / 256, 256, 0, stream>>>(f_dee, 1.0f, N_EDGES);  // self loops

  k_count<<<(NNZ + 255) / 256, 256, 0, stream>>>(src, f_d, NNZ);
  k_count<<<(NNZ + 255) / 256, 256, 0, stream>>>(he, f_b, NNZ);
  k_count<<<(EE_NNZ + 255) / 256, 256, 0, stream>>>(ee_col, f_dee, EE_NNZ);

  // ---- stage 1: HypergraphConv on nodes ----
  // x = node_embeds @ W_hg
  gemm_wmma_staged<0, 4><<<dim3(1, N_NODES / 16), 128, 0, stream>>>(
      node_embeds, W_hg, nullptr, s_x);
  // out_e = B^-1 * scatter(x[src] -> he)
  k_scatter_table<<<NNZ, C1, 0, stream>>>(s_x, src, he, bufA, C1);
  k_scale_rows<<<N_EDGES, C1, 0, stream>>>(bufA, f_b, C1);
  // y = leaky(D^-1 * scatter(out_e[he] -> src) + b_hg)
  k_gather_nodes<<<NNZ, C1, 0, stream>>>(bufA, src, he, s_y, C1);
  k_node_epi<<<N_NODES, C1, 0, stream>>>(s_y, f_d, b_hg, C1);
  // yty = y^T @ y ; z = leaky(yty @ W_lin + b_lin)
  gemm_wmma_f32<0, true><<<dim3(C1 / 16, C1 / 16), 32, 0, stream>>>(
      s_y, s_y, nullptr, s_yty, C1, N_NODES, C1);
  gemm_wmma_staged<1, 4><<<dim3(1, C1 / 16), 128, 0, stream>>>(
      s_yty, W_lin, b_lin, s_z);

  // ---- stage 2: mean-aggregate to hyperedges, GCNConv ----
  // y_edge = scatter(z[src] -> he) / max(cnt,1)
  k_scatter_table<<<NNZ, C1, 0, stream>>>(s_z, src, he, bufB, C1);
  k_scale_rows<<<N_EDGES, C1, 0, stream>>>(bufB, f_b, C1);
  // xw = y_edge @ W_gcn   (bufA fully overwritten)
  gemm_wmma_staged<0, 4><<<dim3(1, N_EDGES / 16), 128, 0, stream>>>(
      bufB, W_gcn, nullptr, bufA);
  // y_gcn = scatter(norm * xw[row] -> col) + self-loop + b_gcn
  hipMemsetAsync(bufB, 0, bigBytes, stream);
  k_gcn_agg<<<EE_NNZ, C2, 0, stream>>>(bufA, ee_row, ee_col, f_dee, bufB, C2);
  k_gcn_epi<<<N_EDGES, C2, 0, stream>>>(bufB, bufA, f_dee, b_gcn, C2);
  // y2e = leaky(y_gcn @ W_lin2 + b_lin2)  (overwrite bufA)
  gemm_wmma_staged<1, 4><<<dim3(1, N_EDGES / 16), 128, 0, stream>>>(
      bufB, W_lin2, b_lin2, bufA);

  // ---- stage 3: motif min + fused affine output ----
  k_motif_min<<<N_MOTIF, C2, 0, stream>>>(bufA, m_edge, bufB, C2);
  // Wc = W_lin3 @ W_out ; bc = b_lin3 @ W_out + b_out
  gemm_wmma_f32<0, false><<<dim3(COUT / 16, C2 / 16), 32, 0, stream>>>(
      W_lin3, W_out, nullptr, s_Wc, C2, C3, COUT);
  k_fuse_bias<<<1, COUT, 0, stream>>>(b_lin3, W_out, b_out, s_bc);
  // out = m @ Wc + bc
  gemm_wmma_staged<0, 1><<<dim3(1, N_MOTIF / 16), 128, 0, stream>>>(
      bufB, s_Wc, s_bc, out);
}